// PCAVolume_4870492914279
// MI455X (gfx1250) — compile-verified
//
#include <hip/hip_runtime.h>
#include <stdint.h>

#define TILE 256
#define SCATTER_BLOCKS 3072

// ---------------------------------------------------------------------------
// Non-returning relaxed agent-scope f32 atomic add -> global_atomic_add_f32
// ---------------------------------------------------------------------------
__device__ __forceinline__ void atomAdd(float* p, float v) {
  __hip_atomic_fetch_add(p, v, __ATOMIC_RELAXED, __HIP_MEMORY_SCOPE_AGENT);
}

// ---------------------------------------------------------------------------
// Workspace zero-init (must run every call; harness does not re-zero ws)
// ---------------------------------------------------------------------------
__global__ __launch_bounds__(256) void zero_ws(float* __restrict__ ws, int n) {
  int i = blockIdx.x * blockDim.x + threadIdx.x;
  int stride = gridDim.x * blockDim.x;
  for (; i < n; i += stride) ws[i] = 0.0f;
}

// ---------------------------------------------------------------------------
// Pass 1: single-pass raw-moment scatter.
// Edge index streams are staged into LDS with gfx1250 async global->LDS DMA,
// double-buffered, so index fetch overlaps the gather+atomic work.
// Per edge: 1 count + 4 coord sums + 6 second moments = 11 f32 atomics (L2).
// ---------------------------------------------------------------------------
__global__ __launch_bounds__(256) void scatter_pass(
    const float4* __restrict__ pts,
    const int* __restrict__ e_base,
    const int* __restrict__ e_voxel,
    float* __restrict__ vol,
    float* __restrict__ sum4,
    float* __restrict__ m2,
    int n_edges, int n_tiles)
{
  __shared__ int sh_b[2][TILE];
  __shared__ int sh_v[2][TILE];
  const int t = threadIdx.x;
  const unsigned long long pb = (unsigned long long)e_base;
  const unsigned long long pv = (unsigned long long)e_voxel;

  int tile = blockIdx.x;

  // prologue: async-stage my first tile into buffer 0
  if (tile < n_tiles) {
    uint32_t e = (uint32_t)(tile * TILE + t);
    if (e >= (uint32_t)n_edges) e = (uint32_t)(n_edges - 1);
    uint32_t off = e * 4u;
    uint32_t lb = (uint32_t)(uintptr_t)&sh_b[0][t];
    uint32_t lv = (uint32_t)(uintptr_t)&sh_v[0][t];
    asm volatile("global_load_async_to_lds_b32 %0, %1, %2"
                 :: "v"(lb), "v"(off), "s"(pb) : "memory");
    asm volatile("global_load_async_to_lds_b32 %0, %1, %2"
                 :: "v"(lv), "v"(off), "s"(pv) : "memory");
  }

  int buf = 0;
  for (; tile < n_tiles; tile += (int)gridDim.x, buf ^= 1) {
    int next = tile + (int)gridDim.x;
    if (next < n_tiles) {
      // issue next tile into the other buffer, then wait for current tile
      uint32_t e = (uint32_t)(next * TILE + t);
      if (e >= (uint32_t)n_edges) e = (uint32_t)(n_edges - 1);
      uint32_t off = e * 4u;
      uint32_t lb = (uint32_t)(uintptr_t)&sh_b[buf ^ 1][t];
      uint32_t lv = (uint32_t)(uintptr_t)&sh_v[buf ^ 1][t];
      asm volatile("global_load_async_to_lds_b32 %0, %1, %2"
                   :: "v"(lb), "v"(off), "s"(pb) : "memory");
      asm volatile("global_load_async_to_lds_b32 %0, %1, %2"
                   :: "v"(lv), "v"(off), "s"(pv) : "memory");
      // 2 newer ops outstanding -> ASYNCcnt<=2 means current tile landed
      asm volatile("s_wait_asynccnt 0x2" ::: "memory");
    } else {
      asm volatile("s_wait_asynccnt 0x0" ::: "memory");
    }
    __syncthreads();   // all 8 waves' slices of this tile are in LDS

    int eidx = tile * TILE + t;
    if (eidx < n_edges) {
      int ib = sh_b[buf][t];
      int iv = sh_v[buf][t];
      float4 p = pts[ib];                 // random 16B gather (L2-resident)
      float x = p.y, y = p.z, z = p.w;    // spatial coords (cols 1:4)
      atomAdd(vol + iv, 1.0f);
      float* s = sum4 + 4 * iv;
      atomAdd(s + 0, p.x); atomAdd(s + 1, x);
      atomAdd(s + 2, y);   atomAdd(s + 3, z);
      float* q = m2 + 6 * iv;
      atomAdd(q + 0, x * x); atomAdd(q + 1, x * y); atomAdd(q + 2, x * z);
      atomAdd(q + 3, y * y); atomAdd(q + 4, y * z); atomAdd(q + 5, z * z);
    }
    __syncthreads();   // buffer `buf` free for the next iteration's prefetch
  }
}

// ---------------------------------------------------------------------------
// 3x3 symmetric eigensolver helpers (closed form, ascending eigenvalues)
// ---------------------------------------------------------------------------
struct F3 { float x, y, z; };
__device__ __forceinline__ F3 xp(F3 a, F3 b) {
  return F3{a.y * b.z - a.z * b.y, a.z * b.x - a.x * b.z, a.x * b.y - a.y * b.x};
}
__device__ __forceinline__ float dt(F3 a, F3 b) { return a.x*b.x + a.y*b.y + a.z*b.z; }

__device__ __forceinline__ bool eigvec3(float a00, float a01, float a02,
                                        float a11, float a12, float a22,
                                        float lam, F3& out) {
  F3 r0{a00 - lam, a01, a02};
  F3 r1{a01, a11 - lam, a12};
  F3 r2{a02, a12, a22 - lam};
  F3 c0 = xp(r0, r1), c1 = xp(r0, r2), c2 = xp(r1, r2);
  float n0 = dt(c0, c0), n1 = dt(c1, c1), n2 = dt(c2, c2);
  F3 best = c0; float nb = n0;
  if (n1 > nb) { best = c1; nb = n1; }
  if (n2 > nb) { best = c2; nb = n2; }
  if (nb < 1e-30f) return false;
  float inv = rsqrtf(nb);
  out = F3{best.x * inv, best.y * inv, best.z * inv};
  return true;
}

__device__ __forceinline__ F3 any_ortho(F3 w) {
  F3 a = (fabsf(w.x) < 0.9f) ? F3{1.f, 0.f, 0.f} : F3{0.f, 1.f, 0.f};
  F3 c = xp(w, a);
  float inv = rsqrtf(fmaxf(dt(c, c), 1e-30f));
  return F3{c.x * inv, c.y * inv, c.z * inv};
}

// ---------------------------------------------------------------------------
// Pass 2: per-voxel finalize. mean / where(mask) / covariance = M2/n - mu mu^T
// then analytic eigh. Outputs packed in reference tuple order.
// ---------------------------------------------------------------------------
__global__ __launch_bounds__(256) void finalize_pass(
    const float* __restrict__ vol,
    const float* __restrict__ sum4,
    const float* __restrict__ m2,
    const float4* __restrict__ bcenter,
    float* __restrict__ out,
    int n_voxels)
{
  int v = blockIdx.x * blockDim.x + threadIdx.x;
  if (v >= n_voxels) return;

  float n = vol[v];
  float denom = fmaxf(n, 1.0f);
  float inv = 1.0f / denom;
  bool mask = n > 0.5f;

  float s0 = sum4[4*v+0] * inv, s1 = sum4[4*v+1] * inv;
  float s2 = sum4[4*v+2] * inv, s3 = sum4[4*v+3] * inv;
  float4 bc = bcenter[v];

  // bxyz
  out[4*v+0] = mask ? s0 : bc.x;
  out[4*v+1] = mask ? s1 : bc.y;
  out[4*v+2] = mask ? s2 : bc.z;
  out[4*v+3] = mask ? s3 : bc.w;
  // volume
  out[4*n_voxels + v] = n;

  // covariance of spatial coords: M2/denom - mu mu^T  (exact for n>=1; 0 for n=0)
  float a00 = m2[6*v+0]*inv - s1*s1;
  float a01 = m2[6*v+1]*inv - s1*s2;
  float a02 = m2[6*v+2]*inv - s1*s3;
  float a11 = m2[6*v+3]*inv - s2*s2;
  float a12 = m2[6*v+4]*inv - s2*s3;
  float a22 = m2[6*v+5]*inv - s3*s3;

  float* ev_out = out + 5*n_voxels + 3*v;
  float* vc_out = out + 8*n_voxels + 9*v;

  float q = (a00 + a11 + a22) * (1.0f/3.0f);
  float p1 = a01*a01 + a02*a02 + a12*a12;
  float d0 = a00 - q, d1 = a11 - q, d2 = a22 - q;
  float p2 = d0*d0 + d1*d1 + d2*d2 + 2.0f * p1;
  float p = sqrtf(p2 * (1.0f/6.0f));

  float e0, e1, e2;
  F3 v0, v1, v2;
  if (p < 1e-12f) {              // (near-)isotropic, incl. empty voxels
    e0 = e1 = e2 = q;
    v0 = F3{1,0,0}; v1 = F3{0,1,0}; v2 = F3{0,0,1};
  } else {
    float ip = 1.0f / p;
    float b00 = d0*ip, b11 = d1*ip, b22 = d2*ip;
    float b01 = a01*ip, b02 = a02*ip, b12 = a12*ip;
    float detB = b00*(b11*b22 - b12*b12) - b01*(b01*b22 - b12*b02)
               + b02*(b01*b12 - b11*b02);
    float r = fminf(fmaxf(0.5f * detB, -1.0f), 1.0f);
    float phi = acosf(r) * (1.0f/3.0f);
    e2 = q + 2.0f*p*cosf(phi);                  // largest
    e0 = q + 2.0f*p*cosf(phi + 2.0943951023931953f); // smallest (+2pi/3)
    e1 = 3.0f*q - e0 - e2;

    bool ok0 = eigvec3(a00,a01,a02,a11,a12,a22, e0, v0);
    bool ok2 = eigvec3(a00,a01,a02,a11,a12,a22, e2, v2);
    if (!ok0 && !ok2)      { v0 = F3{1,0,0}; v2 = F3{0,0,1}; }
    else if (!ok0)         { v0 = any_ortho(v2); }
    else if (!ok2)         { v2 = any_ortho(v0); }
    // orthogonalize v2 against v0, then complete the frame
    float dp = dt(v2, v0);
    F3 tv{v2.x - dp*v0.x, v2.y - dp*v0.y, v2.z - dp*v0.z};
    float tn = dt(tv, tv);
    if (tn > 1e-24f) {
      float ti = rsqrtf(tn);
      v2 = F3{tv.x*ti, tv.y*ti, tv.z*ti};
    } else {
      v2 = any_ortho(v0);
    }
    v1 = xp(v2, v0);
  }

  ev_out[0] = e0; ev_out[1] = e1; ev_out[2] = e2;
  // eigvecs[:, :, i] is i-th eigenvector -> store as columns, row-major [3][3]
  vc_out[0] = v0.x; vc_out[1] = v1.x; vc_out[2] = v2.x;
  vc_out[3] = v0.y; vc_out[4] = v1.y; vc_out[5] = v2.y;
  vc_out[6] = v0.z; vc_out[7] = v1.z; vc_out[8] = v2.z;
}

// ---------------------------------------------------------------------------
extern "C" void kernel_launch(void* const* d_in, const int* in_sizes, int n_in,
                              void* d_out, int out_size, void* d_ws, size_t ws_size,
                              hipStream_t stream) {
  const float4* base_bxyz = (const float4*)d_in[0];  // [P,4] f32
  const float4* bcenter   = (const float4*)d_in[1];  // [V,4] f32
  const int*    e_base    = (const int*)d_in[2];     // [E] int32
  const int*    e_voxel   = (const int*)d_in[3];     // [E] int32
  float* out = (float*)d_out;
  float* ws  = (float*)d_ws;

  const int n_voxels = in_sizes[1] / 4;
  const int n_edges  = in_sizes[2];

  float* vol  = ws;                  // [V]
  float* sum4 = ws + n_voxels;       // [V,4]
  float* m2   = ws + 5 * n_voxels;   // [V,6] xx,xy,xz,yy,yz,zz

  zero_ws<<<1024, 256, 0, stream>>>(ws, 11 * n_voxels);

  int n_tiles = (n_edges + TILE - 1) / TILE;
  int blocks = SCATTER_BLOCKS < n_tiles ? SCATTER_BLOCKS : n_tiles;
  scatter_pass<<<blocks, TILE, 0, stream>>>(base_bxyz, e_base, e_voxel,
                                            vol, sum4, m2, n_edges, n_tiles);

  finalize_pass<<<(n_voxels + 255) / 256, 256, 0, stream>>>(
      vol, sum4, m2, bcenter, out, n_voxels);
}